// HAN_45191645888535
// MI455X (gfx1250) — compile-verified
//
#include <hip/hip_runtime.h>
#include <math.h>

typedef __attribute__((ext_vector_type(2))) float v2f;
typedef __attribute__((ext_vector_type(8))) float v8f;

#define HAN_NA 50000
#define HAN_NT 100000
#define HAN_HID 128
#define HAN_OUT 64
#define HAN_HEADS 8
#define HAN_E1 500000
#define HAN_E2 250000
#define HAN_NEG 0.2f

// ---------------------------------------------------------------- utilities

__device__ __forceinline__ void atomicMaxF(float* addr, float val) {
  int* ai = (int*)addr;
  int old = *ai;
  while (__int_as_float(old) < val) {
    int assumed = old;
    old = atomicCAS(ai, assumed, __float_as_int(val));
    if (old == assumed) break;
  }
}

__global__ void han_fill(float* __restrict__ p, float v, long long n) {
  long long i = (long long)blockIdx.x * blockDim.x + threadIdx.x;
  if (i < n) p[i] = v;
}

__global__ void han_relu_ip(float* __restrict__ p, long long n) {
  long long i = (long long)blockIdx.x * blockDim.x + threadIdx.x;
  if (i < n) p[i] = fmaxf(p[i], 0.0f);
}

// ------------------------------------------------------------ WMMA GEMMs
// C[M,N] = A[M,K] @ B[K,N] + bias[N], all f32 row-major.
// One wave (32 lanes) per 16x16 tile; V_WMMA_F32_16X16X4_F32, K step 4.
// A frag layout: lane L (L<16): row M=L, v={K0,K1}; L>=16: row M=L-16, v={K2,K3}.
// B frag layout symmetric (row K in half-vgpr across 16 lanes of N).
// C/D: vgpr r -> row r (lanes 0-15) / row r+8 (lanes 16-31), col = lane&15.

__global__ __launch_bounds__(32) void han_gemm_wmma(
    const float* __restrict__ A, const float* __restrict__ B,
    const float* __restrict__ bias, float* __restrict__ C,
    int M, int N, int K) {
  const int ntn = N >> 4;
  const int tm = blockIdx.x / ntn;
  const int tn = blockIdx.x - tm * ntn;
  const int lane = threadIdx.x;
  const int half = lane >> 4;
  const int l16 = lane & 15;
  const int arow = tm * 16 + l16;
  const int bcol = tn * 16 + l16;
  const float* ap = A + (size_t)arow * K;
  v8f acc = {0.f, 0.f, 0.f, 0.f, 0.f, 0.f, 0.f, 0.f};
  for (int k0 = 0; k0 < K; k0 += 4) {
    const int ka = k0 + half * 2;
    v2f a, b;
    a.x = ap[ka];
    a.y = ap[ka + 1];
    b.x = B[(size_t)ka * N + bcol];
    b.y = B[(size_t)(ka + 1) * N + bcol];
    acc = __builtin_amdgcn_wmma_f32_16x16x4_f32(false, a, false, b,
                                                (short)0, acc, false, false);
  }
  const float bv = bias[bcol];
#pragma unroll
  for (int r = 0; r < 8; ++r) {
    const int row = tm * 16 + r + half * 8;
    C[(size_t)row * N + bcol] = acc[r] + bv;
  }
}

// Fused semantic-attention score:  *score += sum_{n,f} tanh((A@B)[n,f]+kb[f])*q[f]
__global__ __launch_bounds__(32) void han_gemm_wmma_score(
    const float* __restrict__ A, const float* __restrict__ B,
    const float* __restrict__ kb, const float* __restrict__ q,
    float* __restrict__ score, int M, int N, int K) {
  const int ntn = N >> 4;
  const int tm = blockIdx.x / ntn;
  const int tn = blockIdx.x - tm * ntn;
  const int lane = threadIdx.x;
  const int half = lane >> 4;
  const int l16 = lane & 15;
  const int arow = tm * 16 + l16;
  const int bcol = tn * 16 + l16;
  const float* ap = A + (size_t)arow * K;
  v8f acc = {0.f, 0.f, 0.f, 0.f, 0.f, 0.f, 0.f, 0.f};
  for (int k0 = 0; k0 < K; k0 += 4) {
    const int ka = k0 + half * 2;
    v2f a, b;
    a.x = ap[ka];
    a.y = ap[ka + 1];
    b.x = B[(size_t)ka * N + bcol];
    b.y = B[(size_t)(ka + 1) * N + bcol];
    acc = __builtin_amdgcn_wmma_f32_16x16x4_f32(false, a, false, b,
                                                (short)0, acc, false, false);
  }
  const float kbv = kb[bcol];
  const float qv = q[bcol];
  float local = 0.f;
#pragma unroll
  for (int r = 0; r < 8; ++r) local += tanhf(acc[r] + kbv) * qv;
  for (int m = 16; m >= 1; m >>= 1) local += __shfl_xor(local, m, 32);
  if (lane == 0) atomicAdd(score, local);
}

// ------------------------------------------------------ graph attention

// out[i] = sum_d z[i*D+d] * att[(i%H)*D+d], i over N*H (z is [N, H*D])
__global__ void han_node_alpha(const float* __restrict__ z,
                               const float* __restrict__ att,
                               float* __restrict__ out, long long NH, int H, int D) {
  long long i = (long long)blockIdx.x * blockDim.x + threadIdx.x;
  if (i >= NH) return;
  const int h = (int)(i % H);
  const float* zp = z + i * D;
  const float* ap = att + h * D;
  float s = 0.f;
  for (int d = 0; d < D; ++d) s += zp[d] * ap[d];
  out[i] = s;
}

__device__ __forceinline__ float han_edge_alpha(const float* as_, const float* ad_,
                                                int s, int d, int h, int H) {
  float a = as_[(size_t)s * H + h] + ad_[(size_t)d * H + h];
  return a > 0.f ? a : HAN_NEG * a;
}

__global__ void han_edge_max(const int* __restrict__ src, const int* __restrict__ dst,
                             const float* __restrict__ as_, const float* __restrict__ ad_,
                             float* __restrict__ m, int E, int H) {
  long long i = (long long)blockIdx.x * blockDim.x + threadIdx.x;
  if (i >= (long long)E * H) return;
  const int e = (int)(i / H), h = (int)(i - (long long)e * H);
  const int s = src[e], d = dst[e];
  atomicMaxF(&m[(size_t)d * H + h], han_edge_alpha(as_, ad_, s, d, h, H));
}

__global__ void han_edge_denom(const int* __restrict__ src, const int* __restrict__ dst,
                               const float* __restrict__ as_, const float* __restrict__ ad_,
                               const float* __restrict__ m, float* __restrict__ den,
                               int E, int H) {
  long long i = (long long)blockIdx.x * blockDim.x + threadIdx.x;
  if (i >= (long long)E * H) return;
  const int e = (int)(i / H), h = (int)(i - (long long)e * H);
  const int s = src[e], d = dst[e];
  const float a = han_edge_alpha(as_, ad_, s, d, h, H);
  atomicAdd(&den[(size_t)d * H + h], expf(a - m[(size_t)d * H + h]));
}

__global__ void han_edge_scatter(const int* __restrict__ src, const int* __restrict__ dst,
                                 const float* __restrict__ as_, const float* __restrict__ ad_,
                                 const float* __restrict__ m, const float* __restrict__ den,
                                 const float* __restrict__ xs, float* __restrict__ out,
                                 int E, int H, int D) {
  const int F = H * D;
  long long i = (long long)blockIdx.x * blockDim.x + threadIdx.x;
  if (i >= (long long)E * F) return;
  const int e = (int)(i / F), c = (int)(i - (long long)e * F);
  const int h = c / D;
  const int s = src[e], d = dst[e];
  const float a = han_edge_alpha(as_, ad_, s, d, h, H);
  const float ex = expf(a - m[(size_t)d * H + h]);
  const float w = ex / fmaxf(den[(size_t)d * H + h], 1e-16f);
  atomicAdd(&out[(size_t)d * F + c], xs[(size_t)s * F + c] * w);
}

// ------------------------------------------------- semantic combine / LN / out

__global__ void han_combine(const float* __restrict__ sc, float invN,
                            const float* __restrict__ o0, const float* __restrict__ o1,
                            float* __restrict__ out, long long n) {
  long long i = (long long)blockIdx.x * blockDim.x + threadIdx.x;
  if (i >= n) return;
  const float s0 = sc[0] * invN, s1 = sc[1] * invN;
  const float mx = fmaxf(s0, s1);
  const float e0 = expf(s0 - mx), e1 = expf(s1 - mx);
  const float inv = 1.f / (e0 + e1);
  out[i] = (e0 * o0[i] + e1 * o1[i]) * inv;
}

// in-place relu(layernorm(x)) over feature dim F (32 or 64 or 128), one wave/node
__global__ __launch_bounds__(32) void han_ln_relu(float* __restrict__ x,
                                                  const float* __restrict__ g,
                                                  const float* __restrict__ b, int F) {
  const int lane = threadIdx.x;
  float* xp = x + (size_t)blockIdx.x * F;
  const int cnt = F >> 5;  // F/32, <=4
  float vals[4];
  float s = 0.f;
  for (int j = 0; j < cnt; ++j) { vals[j] = xp[j * 32 + lane]; s += vals[j]; }
  for (int m = 16; m >= 1; m >>= 1) s += __shfl_xor(s, m, 32);
  const float mu = s / (float)F;
  float vs = 0.f;
  for (int j = 0; j < cnt; ++j) { const float d = vals[j] - mu; vs += d * d; }
  for (int m = 16; m >= 1; m >>= 1) vs += __shfl_xor(vs, m, 32);
  const float rstd = rsqrtf(vs / (float)F + 1e-5f);
  for (int j = 0; j < cnt; ++j) {
    const int f = j * 32 + lane;
    xp[f] = fmaxf((vals[j] - mu) * rstd * g[f] + b[f], 0.f);
  }
}

__global__ void han_final_linear(const float* __restrict__ x, const float* __restrict__ W,
                                 const float* __restrict__ b, float* __restrict__ out,
                                 int M, int K, int N) {
  long long i = (long long)blockIdx.x * blockDim.x + threadIdx.x;
  if (i >= (long long)M * N) return;
  const int m = (int)(i / N), c = (int)(i - (long long)m * N);
  const float* xp = x + (size_t)m * K;
  float acc = b[c];
  for (int k = 0; k < K; ++k) acc += xp[k] * W[(size_t)k * N + c];
  out[i] = acc;
}

// ---------------------------------------------------------------- driver

static inline unsigned han_blk(long long n) { return (unsigned)((n + 255) / 256); }

extern "C" void kernel_launch(void* const* d_in, const int* in_sizes, int n_in,
                              void* d_out, int out_size, void* d_ws, size_t ws_size,
                              hipStream_t stream) {
  (void)in_sizes; (void)n_in; (void)out_size; (void)ws_size;
  const float* x_address     = (const float*)d_in[0];
  const float* x_transaction = (const float*)d_in[1];
  const int* eat_src = (const int*)d_in[2];  const int* eat_dst = (const int*)d_in[3];
  const int* eta_src = (const int*)d_in[4];  const int* eta_dst = (const int*)d_in[5];
  const int* eaa_src = (const int*)d_in[6];  const int* eaa_dst = (const int*)d_in[7];
  const int* ett_src = (const int*)d_in[8];  const int* ett_dst = (const int*)d_in[9];
  const float* W_a1 = (const float*)d_in[10]; const float* b_a1 = (const float*)d_in[11];
  const float* W_t1 = (const float*)d_in[12]; const float* b_t1 = (const float*)d_in[13];
  const float* as_at1 = (const float*)d_in[14]; const float* ad_at1 = (const float*)d_in[15];
  const float* as_ta1 = (const float*)d_in[16]; const float* ad_ta1 = (const float*)d_in[17];
  const float* as_aa1 = (const float*)d_in[18]; const float* ad_aa1 = (const float*)d_in[19];
  const float* as_tt1 = (const float*)d_in[20]; const float* ad_tt1 = (const float*)d_in[21];
  const float* k1_W = (const float*)d_in[22]; const float* k1_b = (const float*)d_in[23];
  const float* q1   = (const float*)d_in[24];
  const float* W_a2 = (const float*)d_in[25]; const float* b_a2 = (const float*)d_in[26];
  const float* W_t2 = (const float*)d_in[27]; const float* b_t2 = (const float*)d_in[28];
  const float* as_at2 = (const float*)d_in[29]; const float* ad_at2 = (const float*)d_in[30];
  const float* as_ta2 = (const float*)d_in[31]; const float* ad_ta2 = (const float*)d_in[32];
  const float* as_aa2 = (const float*)d_in[33]; const float* ad_aa2 = (const float*)d_in[34];
  const float* as_tt2 = (const float*)d_in[35]; const float* ad_tt2 = (const float*)d_in[36];
  const float* k2_W = (const float*)d_in[37]; const float* k2_b = (const float*)d_in[38];
  const float* q2   = (const float*)d_in[39];
  const float* ln1_g = (const float*)d_in[40]; const float* ln1_b = (const float*)d_in[41];
  const float* ln2_g = (const float*)d_in[42]; const float* ln2_b = (const float*)d_in[43];
  const float* lin_W = (const float*)d_in[44]; const float* lin_b = (const float*)d_in[45];
  float* out = (float*)d_out;

  // ---- workspace layout (floats) ----
  float* w = (float*)d_ws;
  size_t off = 0;
  auto alloc = [&](size_t n) { float* p = w + off; off += n; return p; };
  float* za   = alloc((size_t)HAN_NA * HAN_HID);   // layer1 proj A / new_a
  float* zt   = alloc((size_t)HAN_NT * HAN_HID);   // layer1 proj T / new_t
  float* za2  = alloc((size_t)HAN_NA * HAN_OUT);   // layer2 proj A / new_a2
  float* zt2  = alloc((size_t)HAN_NT * HAN_OUT);   // layer2 proj T
  float* o_at = alloc((size_t)HAN_NT * HAN_HID);
  float* o_ta = alloc((size_t)HAN_NA * HAN_HID);
  float* o_aa = alloc((size_t)HAN_NA * HAN_HID);
  float* o_tt = alloc((size_t)HAN_NT * HAN_HID);
  float* alpS = alloc((size_t)HAN_NT * HAN_HEADS);
  float* alpD = alloc((size_t)HAN_NT * HAN_HEADS);
  float* mbuf = alloc((size_t)HAN_NT * HAN_HEADS);
  float* dbuf = alloc((size_t)HAN_NT * HAN_HEADS);
  float* score = alloc(8);

  auto gemm = [&](const float* A, const float* B, const float* bias, float* C,
                  int M, int N, int K) {
    han_gemm_wmma<<<dim3((unsigned)((M / 16) * (N / 16))), 32, 0, stream>>>(A, B, bias, C, M, N, K);
  };
  auto gemm_score = [&](const float* A, const float* B, const float* kb, const float* q,
                        float* sc, int M, int N, int K) {
    han_gemm_wmma_score<<<dim3((unsigned)((M / 16) * (N / 16))), 32, 0, stream>>>(A, B, kb, q, sc, M, N, K);
  };
  auto run_edge = [&](const int* src, const int* dst, int E,
                      const float* zsrc, const float* zdst,
                      const float* att_s, const float* att_d,
                      int Nsrc, int Ndst, int H, int D, float* o) {
    const int F = H * D;
    const long long nsh = (long long)Nsrc * H, ndh = (long long)Ndst * H;
    const long long ndf = (long long)Ndst * F;
    han_node_alpha<<<han_blk(nsh), 256, 0, stream>>>(zsrc, att_s, alpS, nsh, H, D);
    han_node_alpha<<<han_blk(ndh), 256, 0, stream>>>(zdst, att_d, alpD, ndh, H, D);
    han_fill<<<han_blk(ndh), 256, 0, stream>>>(mbuf, -INFINITY, ndh);
    han_fill<<<han_blk(ndh), 256, 0, stream>>>(dbuf, 0.f, ndh);
    han_fill<<<han_blk(ndf), 256, 0, stream>>>(o, 0.f, ndf);
    han_edge_max<<<han_blk((long long)E * H), 256, 0, stream>>>(src, dst, alpS, alpD, mbuf, E, H);
    han_edge_denom<<<han_blk((long long)E * H), 256, 0, stream>>>(src, dst, alpS, alpD, mbuf, dbuf, E, H);
    han_edge_scatter<<<han_blk((long long)E * F), 256, 0, stream>>>(src, dst, alpS, alpD, mbuf, dbuf,
                                                                    zsrc, o, E, H, D);
    han_relu_ip<<<han_blk(ndf), 256, 0, stream>>>(o, ndf);  // HANConv relu per edge type
  };

  // ================= Layer 1 (HID=128, D=16) =================
  gemm(x_address, W_a1, b_a1, za, HAN_NA, HAN_HID, 128);
  gemm(x_transaction, W_t1, b_t1, zt, HAN_NT, HAN_HID, 64);

  run_edge(eat_src, eat_dst, HAN_E1, za, zt, as_at1, ad_at1, HAN_NA, HAN_NT, HAN_HEADS, 16, o_at);
  run_edge(eta_src, eta_dst, HAN_E1, zt, za, as_ta1, ad_ta1, HAN_NT, HAN_NA, HAN_HEADS, 16, o_ta);
  run_edge(eaa_src, eaa_dst, HAN_E2, za, za, as_aa1, ad_aa1, HAN_NA, HAN_NA, HAN_HEADS, 16, o_aa);
  run_edge(ett_src, ett_dst, HAN_E2, zt, zt, as_tt1, ad_tt1, HAN_NT, HAN_NT, HAN_HEADS, 16, o_tt);

  // semantic attention (K=2 metapaths per node type)
  han_fill<<<1, 256, 0, stream>>>(score, 0.f, 8);
  gemm_score(o_ta, k1_W, k1_b, q1, &score[0], HAN_NA, HAN_HID, HAN_HID);
  gemm_score(o_aa, k1_W, k1_b, q1, &score[1], HAN_NA, HAN_HID, HAN_HID);
  gemm_score(o_at, k1_W, k1_b, q1, &score[2], HAN_NT, HAN_HID, HAN_HID);
  gemm_score(o_tt, k1_W, k1_b, q1, &score[3], HAN_NT, HAN_HID, HAN_HID);
  han_combine<<<han_blk((long long)HAN_NA * HAN_HID), 256, 0, stream>>>(
      &score[0], 1.0f / HAN_NA, o_ta, o_aa, za, (long long)HAN_NA * HAN_HID);
  han_combine<<<han_blk((long long)HAN_NT * HAN_HID), 256, 0, stream>>>(
      &score[2], 1.0f / HAN_NT, o_at, o_tt, zt, (long long)HAN_NT * HAN_HID);

  han_ln_relu<<<dim3(HAN_NA), 32, 0, stream>>>(za, ln1_g, ln1_b, HAN_HID);
  han_ln_relu<<<dim3(HAN_NT), 32, 0, stream>>>(zt, ln1_g, ln1_b, HAN_HID);

  // ================= Layer 2 (OUT=64, D=8) =================
  gemm(za, W_a2, b_a2, za2, HAN_NA, HAN_OUT, HAN_HID);
  gemm(zt, W_t2, b_t2, zt2, HAN_NT, HAN_OUT, HAN_HID);

  // Only the address branch reaches the output: skip dead at/tt work in layer 2.
  run_edge(eta_src, eta_dst, HAN_E1, zt2, za2, as_ta2, ad_ta2, HAN_NT, HAN_NA, HAN_HEADS, 8, o_ta);
  run_edge(eaa_src, eaa_dst, HAN_E2, za2, za2, as_aa2, ad_aa2, HAN_NA, HAN_NA, HAN_HEADS, 8, o_aa);
  (void)as_at2; (void)ad_at2; (void)as_tt2; (void)ad_tt2;

  han_fill<<<1, 256, 0, stream>>>(score, 0.f, 8);
  gemm_score(o_ta, k2_W, k2_b, q2, &score[0], HAN_NA, HAN_OUT, HAN_OUT);
  gemm_score(o_aa, k2_W, k2_b, q2, &score[1], HAN_NA, HAN_OUT, HAN_OUT);
  han_combine<<<han_blk((long long)HAN_NA * HAN_OUT), 256, 0, stream>>>(
      &score[0], 1.0f / HAN_NA, o_ta, o_aa, za2, (long long)HAN_NA * HAN_OUT);

  han_ln_relu<<<dim3(HAN_NA), 32, 0, stream>>>(za2, ln2_g, ln2_b, HAN_OUT);

  // final classifier: [NA,64] @ [64,2] + b
  han_final_linear<<<han_blk((long long)HAN_NA * 2), 256, 0, stream>>>(
      za2, lin_W, lin_b, out, HAN_NA, HAN_OUT, 2);
}